// PolynomialExpansion_11596411699377
// MI455X (gfx1250) — compile-verified
//
#include <hip/hip_runtime.h>

typedef __attribute__((ext_vector_type(2))) float v2f;
typedef __attribute__((ext_vector_type(8))) float v8f;

#define NF   12          // input features
#define NY   13          // [1, x]
#define NEXP 454         // unique degree-3 monomials
#define NTILE 29         // 29*16 = 464 padded features
#define ROWS_PER_BLOCK 256
#define STATS_BLOCKS_MAX 256
#define EXP_ROWS 8
#define BN_EPS 1e-5f

// ---------------------------------------------------------------------------
// Build monomial triple table (lex order of nondecreasing (c0,c1,c2), skip 0,0,0)
// Matches np.nonzero(mask.ravel()) order in the reference.
// ---------------------------------------------------------------------------
__global__ void init_table(int* __restrict__ tbl) {
  if (threadIdx.x == 0 && blockIdx.x == 0) {
    int q = 0;
    for (int c0 = 0; c0 < NY; ++c0)
      for (int c1 = c0; c1 < NY; ++c1)
        for (int c2 = c1; c2 < NY; ++c2) {
          if ((c0 | c1 | c2) == 0) continue;
          tbl[q++] = c0 | (c1 << 8) | (c2 << 16);
        }
    const int last = tbl[NEXP - 1];
    for (int i = NEXP; i < NTILE * 16; ++i) tbl[i] = last;  // pad (excluded later)
  }
}

// ---------------------------------------------------------------------------
// Pass 1: per-feature Sum(f) and Sum(f^2) via V_WMMA_F32_16X16X4_F32.
// A(16x4) = 4 batch rows x 16 features of monomials (documented A layout:
// lane L holds M=L&15; K = 2*(L>>4) + {vgpr0:0, vgpr1:1}).
// B(4x16) = ones (layout independent). D accumulates column sums, replicated
// across N; extracted from lanes 0 (M=0..7) and 16 (M=8..15).
// Deterministic: fixed-order block partials, no float atomics.
// ---------------------------------------------------------------------------
__global__ __launch_bounds__(256) void stats_kernel(
    const float* __restrict__ x, const int* __restrict__ tbl,
    float* __restrict__ part, int B, int nChunks) {
  __shared__ float y_s[ROWS_PER_BLOCK * NY];
  __shared__ float red_s[8][32];
  __shared__ float part_s[2 * NEXP];

  const int tid  = threadIdx.x;
  const int wave = tid >> 5;
  const int lane = tid & 31;
  const int hi   = lane >> 4;   // 0: K=0,1   1: K=2,3
  const int m    = lane & 15;   // feature-in-tile (A-matrix M)

  for (int i = tid; i < 2 * NEXP; i += 256) part_s[i] = 0.f;

  const v2f ones = {1.0f, 1.0f};

  for (int chunk = blockIdx.x; chunk < nChunks; chunk += gridDim.x) {
    const int rowBase = chunk * ROWS_PER_BLOCK;
    __syncthreads();  // y_s reuse + part_s init visibility
    for (int i = tid; i < ROWS_PER_BLOCK * NY; i += 256) {
      const int r = i / NY, c = i - r * NY;
      const int row = rowBase + r;
      float v = 1.0f;
      if (c != 0) {
        const int rr = row < B ? row : (B - 1);
        v = x[rr * NF + (c - 1)];
      }
      y_s[i] = v;
    }
    __syncthreads();

    const int wrow = wave * 32;  // 8 waves x 32 rows = 256 rows/chunk
    for (int t = 0; t < NTILE; ++t) {
      const int f  = t * 16 + m;
      const int fc = f < NEXP ? f : (NEXP - 1);
      const int trip = tbl[fc];
      const int c0 = trip & 255, c1 = (trip >> 8) & 255, c2 = (trip >> 16) & 255;

      v8f accS = {};
      v8f accQ = {};
      for (int it = 0; it < 8; ++it) {
        const int r0 = wrow + it * 4 + 2 * hi;
        const float* y0 = &y_s[r0 * NY];
        const float* y1 = y0 + NY;
        float v0 = y0[c0] * y0[c1] * y0[c2];
        float v1 = y1[c0] * y1[c1] * y1[c2];
        v0 *= (rowBase + r0     < B) ? 1.f : 0.f;   // arithmetic mask, no EXEC change
        v1 *= (rowBase + r0 + 1 < B) ? 1.f : 0.f;
        v2f a  = {v0, v1};
        v2f aq = {v0 * v0, v1 * v1};
        accS = __builtin_amdgcn_wmma_f32_16x16x4_f32(false, a,  false, ones,
                                                     (short)0, accS, false, false);
        accQ = __builtin_amdgcn_wmma_f32_16x16x4_f32(false, aq, false, ones,
                                                     (short)0, accQ, false, false);
      }
      // D layout: lane0 vgpr r -> (M=r, N=0); lane16 vgpr r -> (M=8+r, N=0)
      if (lane == 0) {
#pragma unroll
        for (int r = 0; r < 8; ++r) { red_s[wave][r]      = accS[r]; red_s[wave][16 + r] = accQ[r]; }
      } else if (lane == 16) {
#pragma unroll
        for (int r = 0; r < 8; ++r) { red_s[wave][8 + r]  = accS[r]; red_s[wave][24 + r] = accQ[r]; }
      }
      __syncthreads();
      if (tid < 32) {
        float s = 0.f;
#pragma unroll
        for (int w = 0; w < 8; ++w) s += red_s[w][tid];
        const int mm = tid & 15;
        const int ff = t * 16 + mm;
        if (ff < NEXP) part_s[(tid < 16 ? 0 : NEXP) + ff] += s;
      }
      __syncthreads();
    }
  }
  __syncthreads();
  for (int i = tid; i < 2 * NEXP; i += 256)
    part[(size_t)blockIdx.x * (2 * NEXP) + i] = part_s[i];
}

// ---------------------------------------------------------------------------
// Fold stats + affine params into per-feature (a, b):  out = f*a + b
// ---------------------------------------------------------------------------
__global__ void finalize_kernel(const float* __restrict__ part,
                                const float* __restrict__ w,
                                const float* __restrict__ bias,
                                float* __restrict__ coefA,
                                float* __restrict__ coefB,
                                int B, int nblk) {
  const int f = blockIdx.x * blockDim.x + threadIdx.x;
  if (f >= NEXP) return;
  float s = 0.f, q = 0.f;
  for (int b = 0; b < nblk; ++b) {
    s += part[(size_t)b * (2 * NEXP) + f];
    q += part[(size_t)b * (2 * NEXP) + NEXP + f];
  }
  const float invB = 1.0f / (float)B;
  const float mean = s * invB;
  float var = q * invB - mean * mean;
  var = fmaxf(var, 0.f);
  const float inv = 1.0f / sqrtf(var + BN_EPS);
  const float a = w[f] * inv;
  coefA[f] = a;
  coefB[f] = bias[f] - mean * a;
}

// ---------------------------------------------------------------------------
// Pass 2: recompute monomials, apply f*a+b, stream out with NT float2 stores.
// 8 rows/block; thread pairs cover the 454 features (454 = 2*227, 8B aligned).
// ---------------------------------------------------------------------------
__global__ __launch_bounds__(256) void expand_kernel(
    const float* __restrict__ x, const int* __restrict__ tbl,
    const float* __restrict__ coefA, const float* __restrict__ coefB,
    float* __restrict__ out, int B) {
  __shared__ float y_s[EXP_ROWS * NY];
  __shared__ int   tbl_s[NEXP];
  __shared__ float a_s[NEXP];
  __shared__ float b_s[NEXP];

  const int tid = threadIdx.x;
  const int rowBase = blockIdx.x * EXP_ROWS;

  for (int i = tid; i < NEXP; i += 256) {
    tbl_s[i] = tbl[i];
    a_s[i]   = coefA[i];
    b_s[i]   = coefB[i];
  }
  for (int i = tid; i < EXP_ROWS * NY; i += 256) {
    const int r = i / NY, c = i - r * NY;
    const int row = rowBase + r;
    float v = 1.0f;
    if (c != 0) {
      const int rr = row < B ? row : (B - 1);
      v = x[rr * NF + (c - 1)];
    }
    y_s[i] = v;
  }
  __syncthreads();

  const int f0 = tid * 2;
  if (f0 < NEXP) {
    const int t0 = tbl_s[f0], t1 = tbl_s[f0 + 1];
    const int p0 = t0 & 255, p1 = (t0 >> 8) & 255, p2 = (t0 >> 16) & 255;
    const int q0 = t1 & 255, q1 = (t1 >> 8) & 255, q2 = (t1 >> 16) & 255;
    const float cA0 = a_s[f0],     cB0 = b_s[f0];
    const float cA1 = a_s[f0 + 1], cB1 = b_s[f0 + 1];
#pragma unroll
    for (int r = 0; r < EXP_ROWS; ++r) {
      const int row = rowBase + r;
      if (row >= B) break;
      const float* y = &y_s[r * NY];
      const float v0 = y[p0] * y[p1] * y[p2];
      const float v1 = y[q0] * y[q1] * y[q2];
      v2f o = {fmaf(v0, cA0, cB0), fmaf(v1, cA1, cB1)};
      __builtin_nontemporal_store(o, (v2f*)(out + (size_t)row * NEXP + f0));
    }
  }
}

// ---------------------------------------------------------------------------
extern "C" void kernel_launch(void* const* d_in, const int* in_sizes, int n_in,
                              void* d_out, int out_size, void* d_ws, size_t ws_size,
                              hipStream_t stream) {
  (void)n_in; (void)out_size;
  const float* x    = (const float*)d_in[0];
  const float* bw   = (const float*)d_in[1];
  const float* bb   = (const float*)d_in[2];
  float*       out  = (float*)d_out;
  const int B = in_sizes[0] / NF;

  // Workspace layout
  char* ws = (char*)d_ws;
  int*   tbl   = (int*)ws;                      // 464 ints   -> pad to 2048 B
  float* coefA = (float*)(ws + 2048);           // 454 floats -> pad
  float* coefB = (float*)(ws + 2048 + 1920);
  float* part  = (float*)(ws + 6144);           // nblk * 908 floats

  const int nChunks = (B + ROWS_PER_BLOCK - 1) / ROWS_PER_BLOCK;
  int nblk = STATS_BLOCKS_MAX;
  if (nblk > nChunks) nblk = nChunks;
  const size_t partCap = (ws_size > 6144) ? (ws_size - 6144) / (2 * NEXP * sizeof(float)) : 1;
  if ((size_t)nblk > partCap) nblk = (int)(partCap ? partCap : 1);

  init_table<<<1, 32, 0, stream>>>(tbl);
  stats_kernel<<<nblk, 256, 0, stream>>>(x, tbl, part, B, nChunks);
  finalize_kernel<<<(NEXP + 127) / 128, 128, 0, stream>>>(part, bw, bb, coefA, coefB, B, nblk);
  const int expBlocks = (B + EXP_ROWS - 1) / EXP_ROWS;
  expand_kernel<<<expBlocks, 256, 0, stream>>>(x, tbl, coefA, coefB, out, B);
}